// HGAT_5025111736685
// MI455X (gfx1250) — compile-verified
//
#include <hip/hip_runtime.h>

typedef __attribute__((ext_vector_type(2))) float v2f;
typedef __attribute__((ext_vector_type(8))) float v8f;

namespace {
constexpr int NB   = 32;    // batch
constexpr int CIN  = 64;    // input channels
constexpr int CREL = 128;   // conv output channels
constexpr int NV   = 1024;  // points
constexpr int NK   = 32;    // top-k

__device__ inline v8f wmma_k4(v2f a, v2f b, v8f c) {
  // D(16x16,f32) = A(16x4,f32) x B(4x16,f32) + C ; full f32 matrix pipe
  return __builtin_amdgcn_wmma_f32_16x16x4_f32(
      /*neg_a=*/false, a, /*neg_b=*/false, b,
      /*c_mod=*/(short)0, c, /*reuse_a=*/false, /*reuse_b=*/false);
}

// ---------------------------------------------------------------------------
// Kernel 1: pre[b] = W(128x64) @ x[b](64x1024) + bias   via V_WMMA_F32_16X16X4
// grid (NV/16, NB), block 256 (8 waves); wave w owns output rows [16w,16w+16)
// ---------------------------------------------------------------------------
__global__ __launch_bounds__(256)
void k_conv(const float* __restrict__ x, const float* __restrict__ W,
            const float* __restrict__ bias, float* __restrict__ pre) {
  const int b    = blockIdx.y;
  const int v0   = blockIdx.x * 16;
  const int wv   = threadIdx.x >> 5;
  const int lane = threadIdx.x & 31;
  const int h    = lane >> 4;       // half-wave: K-pair select
  const int ln   = lane & 15;       // row (A) / col (B) within tile
  const int o0   = wv * 16;

  v8f acc = {};
  const float* wr = W + (size_t)(o0 + ln) * CIN + 2 * h;           // A: W[o, c]
  const float* xc = x + (size_t)b * CIN * NV + (size_t)(2 * h) * NV + v0 + ln;
  for (int kb = 0; kb < CIN; kb += 4) {
    v2f av = { wr[kb], wr[kb + 1] };
    v2f bvv = { xc[(size_t)kb * NV], xc[(size_t)(kb + 1) * NV] };
    acc = wmma_k4(av, bvv, acc);
  }
  float* pb = pre + (size_t)b * CREL * NV;
#pragma unroll
  for (int r = 0; r < 8; ++r) {
    const int row = o0 + r + 8 * h;                // C/D layout: M = r + 8*(lane/16)
    pb[(size_t)row * NV + v0 + ln] = acc[r] + bias[row];
  }
}

// ---------------------------------------------------------------------------
// Kernel 2: xx[b,v] = sum_c pre^2 ; p1 = pre . a[:128] ; p2 = pre . a[128:]
// ---------------------------------------------------------------------------
__global__ __launch_bounds__(256)
void k_stats(const float* __restrict__ pre, const float* __restrict__ a,
             float* __restrict__ xx, float* __restrict__ p1,
             float* __restrict__ p2) {
  const int t = blockIdx.x * blockDim.x + threadIdx.x;   // 0..NB*NV-1
  const int b = t >> 10;
  const int v = t & (NV - 1);
  const float* pb = pre + (size_t)b * CREL * NV + v;
  float sxx = 0.f, s1 = 0.f, s2 = 0.f;
#pragma unroll 8
  for (int c = 0; c < CREL; ++c) {
    const float val = pb[(size_t)c * NV];
    sxx += val * val;
    s1  += val * a[c];
    s2  += val * a[CREL + c];
  }
  xx[t] = sxx; p1[t] = s1; p2[t] = s2;
}

// ---------------------------------------------------------------------------
// Kernel 3 (fused, dominant): per (b, 16-row v-tile):
//   scores[v,u] = 2*(pre_v . pre_u) - xx[v] - xx[u]  via WMMA into 64KB LDS,
//   then wave w extracts the top-32 (desc, tie->lower idx) of row w and emits
//   s[b,v,k] = p1[b,(v*32+k)%1024] + p2[b, idx].  Avoids the 134MB pairwise
//   matrix ever touching HBM.
// grid (NV/16, NB), block 512 (16 waves)
// ---------------------------------------------------------------------------
__global__ __launch_bounds__(512)
void k_gram_topk(const float* __restrict__ pre, const float* __restrict__ xx,
                 const float* __restrict__ p1, const float* __restrict__ p2,
                 float* __restrict__ s_out) {
  __shared__ float Sld[16][NV];                      // 64 KB score strip
  const int b    = blockIdx.y;
  const int v0   = blockIdx.x * 16;
  const int wv   = threadIdx.x >> 5;                 // 0..15
  const int lane = threadIdx.x & 31;
  const int h    = lane >> 4;
  const int ln   = lane & 15;

  const float* pb  = pre + (size_t)b * CREL * NV;
  const float* xxb = xx + b * NV;

  // --- compute: wave wv fills columns [64*wv, 64*wv+64) of the 16x1024 strip
  for (int t = 0; t < 4; ++t) {
    const int u0 = wv * 64 + t * 16;
    v8f acc = {};
    const float* pv = pb + (size_t)(2 * h) * NV;
    for (int kb = 0; kb < CREL; kb += 4) {
      const float* pr = pv + (size_t)kb * NV;
      v2f av  = { pr[v0 + ln], pr[NV + v0 + ln] };   // A = pre^T tile (16x4)
      v2f bvv = { pr[u0 + ln], pr[NV + u0 + ln] };   // B = pre tile  (4x16)
      acc = wmma_k4(av, bvv, acc);
    }
    const float xu = xxb[u0 + ln];
#pragma unroll
    for (int r = 0; r < 8; ++r) {
      const int M = r + 8 * h;
      Sld[M][u0 + ln] = 2.0f * acc[r] - xxb[v0 + M] - xu;
    }
  }
  __syncthreads();

  // --- select: wave wv owns row wv; lane l caches scores [32l, 32l+32)
  float vals[32];
#pragma unroll
  for (int j = 0; j < 32; ++j) vals[j] = Sld[wv][lane * 32 + j];

  const float NEG = -__builtin_inff();
  int myIdx = 0;                                     // lane k keeps k-th winner
  for (int k = 0; k < NK; ++k) {
    float bv = NEG; int bi = 0;
#pragma unroll
    for (int j = 0; j < 32; ++j)                     // strict > keeps lowest idx
      if (vals[j] > bv) { bv = vals[j]; bi = j; }
    int gi = (lane << 5) | bi;
#pragma unroll
    for (int m = 16; m >= 1; m >>= 1) {              // butterfly argmax
      const float ov = __shfl_xor(bv, m, 32);
      const int   og = __shfl_xor(gi, m, 32);
      if (ov > bv || (ov == bv && og < gi)) { bv = ov; gi = og; }
    }
    if (lane == k) myIdx = gi;
    if ((gi >> 5) == lane) {                         // retire winner (reg-resident)
      const int cj = gi & 31;
#pragma unroll
      for (int j = 0; j < 32; ++j) if (j == cj) vals[j] = NEG;
    }
  }

  const int v    = v0 + wv;
  const int jrep = (v * NK + lane) & (NV - 1);       // reference's tile-mod-V wrap
  const float sval = p1[b * NV + jrep] + p2[b * NV + myIdx];
  s_out[(size_t)b * NV * NK + (size_t)v * NK + lane] = sval;
}

// ---------------------------------------------------------------------------
// Kernel 4: softmax over the BATCH axis, in place on d_out. One thread per
// (v,k) column of 32 values; coalesced across threads.
// ---------------------------------------------------------------------------
__global__ __launch_bounds__(256)
void k_softmax_b(float* __restrict__ s) {
  const int col = blockIdx.x * blockDim.x + threadIdx.x;  // 0..NV*NK-1
  float e[NB];
  float m = -__builtin_inff();
#pragma unroll
  for (int b = 0; b < NB; ++b) {
    const float v = s[(size_t)b * NV * NK + col];
    e[b] = v;
    m = fmaxf(m, v);
  }
  float sum = 0.f;
#pragma unroll
  for (int b = 0; b < NB; ++b) { e[b] = __expf(e[b] - m); sum += e[b]; }
  const float inv = 1.0f / sum;
#pragma unroll
  for (int b = 0; b < NB; ++b)
    s[(size_t)b * NV * NK + col] = e[b] * inv;
}

}  // namespace

extern "C" void kernel_launch(void* const* d_in, const int* in_sizes, int n_in,
                              void* d_out, int out_size, void* d_ws,
                              size_t ws_size, hipStream_t stream) {
  (void)in_sizes; (void)n_in; (void)out_size; (void)ws_size;
  const float* x  = (const float*)d_in[0];   // (32, 64, 1024)
  const float* W  = (const float*)d_in[1];   // (128, 64)
  const float* bc = (const float*)d_in[2];   // (128,)
  const float* a  = (const float*)d_in[3];   // (256, 1)
  float* out = (float*)d_out;                // (32, 1024, 32)

  float* pre = (float*)d_ws;                        // 16 MB
  float* xx  = pre + (size_t)NB * CREL * NV;        // +128 KB
  float* p1  = xx + (size_t)NB * NV;                // +128 KB
  float* p2  = p1 + (size_t)NB * NV;                // +128 KB

  k_conv<<<dim3(NV / 16, NB), 256, 0, stream>>>(x, W, bc, pre);
  k_stats<<<dim3(NB * NV / 256), 256, 0, stream>>>(pre, a, xx, p1, p2);
  k_gram_topk<<<dim3(NV / 16, NB), 512, 0, stream>>>(pre, xx, p1, p2, out);
  k_softmax_b<<<dim3(NV * NK / 256), 256, 0, stream>>>(out);
}